// GNNModel_58858231824523
// MI455X (gfx1250) — compile-verified
//
#include <hip/hip_runtime.h>
#include <math.h>

#define H 128
#define NNODE 100000
#define NB 1024
#define NPER 32
#define NTOT (NB * NPER)   /* 32768 */
#define NE (2 * NTOT)      /* 65536 */

typedef float v2f __attribute__((ext_vector_type(2)));
typedef float v8f __attribute__((ext_vector_type(8)));

__device__ __forceinline__ float sigmoidf_(float x) { return 1.0f / (1.0f + expf(-x)); }

// ---------------------------------------------------------------------------
// Generic fp32 WMMA GEMM: D[M,N] = A[M,K] * B + bias(optional, per column)
//   TRANS_B == false : B is K x N row-major (ldb = N stride)
//   TRANS_B == true  : B is N x K row-major (computes A @ B^T), ldb = K stride
// One wave computes one 16x16 tile, K-loop in steps of 4 using
// v_wmma_f32_16x16x4_f32. 8 waves (256 threads) per block.
// Requires M%16==0, N%16==0, K%4==0.
// ---------------------------------------------------------------------------
template <bool TRANS_B>
__global__ void gemm_wmma_f32(const float* __restrict__ A, const float* __restrict__ B,
                              const float* __restrict__ bias, float* __restrict__ D,
                              int M, int N, int K, int lda, int ldb, int ldd) {
    const int lane = threadIdx.x & 31;
    const int wave = threadIdx.x >> 5;
    const int tilesN = N >> 4;
    const int tilesM = M >> 4;
    const int tile = blockIdx.x * 8 + wave;
    if (tile >= tilesM * tilesN) return;  // wave-uniform: EXEC all-ones below
    const int tm = (tile / tilesN) << 4;
    const int tn = (tile % tilesN) << 4;
    const int half = lane >> 4;   // selects K pair (A/B) and M half (C/D)
    const int i16 = lane & 15;    // A: row m ; B/D: col n

    v8f acc = {};
    const float* Arow = A + (long)(tm + i16) * lda;
    const float* Brow = TRANS_B ? (B + (long)(tn + i16) * ldb) : nullptr;
    for (int k0 = 0; k0 < K; k0 += 4) {
        const int ka = k0 + half * 2;
        v2f a, b;
        a.x = Arow[ka];
        a.y = Arow[ka + 1];
        if (TRANS_B) {
            b.x = Brow[ka];
            b.y = Brow[ka + 1];
        } else {
            b.x = B[(long)ka * ldb + tn + i16];
            b.y = B[(long)(ka + 1) * ldb + tn + i16];
        }
        acc = __builtin_amdgcn_wmma_f32_16x16x4_f32(false, a, false, b,
                                                    (short)0, acc, false, false);
    }
    const float bv = bias ? bias[tn + i16] : 0.0f;
    const int rbase = half * 8;
#pragma unroll
    for (int r = 0; r < 8; ++r) {
        D[(long)(tm + rbase + r) * ldd + tn + i16] = acc[r] + bv;
    }
}

// ---------------------------------------------------------------------------
// Specialized scoring GEMM: out[NB, NNODE] = s_h[NB, H] @ emb[NNODE, H]^T
// K = H = 128 fixed. Each wave computes a 16 x (5*16 = 80) super-tile:
//  - 16x128 A strip preloaded once into 32 v2f registers (reused for all 5
//    N-tiles and all 32 K-steps -> A traffic cut 5x, addr math amortized)
//  - inner step: 5 b64 B-loads + 5 WMMAs
// N tiles = NNODE/16 = 6250 = 5 * 1250  (exact, no edge handling; EXEC all-1s)
// ---------------------------------------------------------------------------
#define SCORE_NT 5
#define SCORE_SUPERN (SCORE_NT * 16)               /* 80 */
#define SCORE_TILES_N (NNODE / SCORE_SUPERN)       /* 1250 */
#define SCORE_WAVES ((NB / 16) * SCORE_TILES_N)    /* 80000 */

__global__ void score_gemm_wmma(const float* __restrict__ A /* s_h [NB,H] */,
                                const float* __restrict__ B /* emb [NNODE,H] */,
                                float* __restrict__ D /* out [NB,NNODE] */) {
    const int lane = threadIdx.x & 31;
    const int wave = threadIdx.x >> 5;
    const int tile = blockIdx.x * 8 + wave;
    if (tile >= SCORE_WAVES) return;              // wave-uniform
    const int tm = (tile / SCORE_TILES_N) << 4;
    const int tn = (tile % SCORE_TILES_N) * SCORE_SUPERN;
    const int half = lane >> 4;
    const int i16 = lane & 15;

    // Preload full 16x128 A strip (per-lane: row tm+i16, K pair half*2 per step)
    const float* Arow = A + (long)(tm + i16) * H;
    v2f areg[H / 4];
#pragma unroll
    for (int kk = 0; kk < H / 4; ++kk) {
        const int ka = kk * 4 + half * 2;
        areg[kk].x = Arow[ka];
        areg[kk].y = Arow[ka + 1];
    }

    // B row pointers: emb rows = output columns (TRANS_B access)
    const float* Brow[SCORE_NT];
#pragma unroll
    for (int t = 0; t < SCORE_NT; ++t) Brow[t] = B + (long)(tn + t * 16 + i16) * H;

    v8f acc[SCORE_NT];
#pragma unroll
    for (int t = 0; t < SCORE_NT; ++t) acc[t] = (v8f){};

    for (int kk = 0; kk < H / 4; ++kk) {
        const int ka = kk * 4 + half * 2;
#pragma unroll
        for (int t = 0; t < SCORE_NT; ++t) {
            v2f b;
            b.x = Brow[t][ka];
            b.y = Brow[t][ka + 1];
            acc[t] = __builtin_amdgcn_wmma_f32_16x16x4_f32(false, areg[kk], false, b,
                                                           (short)0, acc[t], false, false);
        }
    }

    const int rbase = half * 8;
#pragma unroll
    for (int t = 0; t < SCORE_NT; ++t) {
#pragma unroll
        for (int r = 0; r < 8; ++r) {
            D[(long)(tm + rbase + r) * NNODE + tn + t * 16 + i16] = acc[t][r];
        }
    }
}

// h[i,:] = emb[x[i]-1, :]
__global__ void gather_emb_kernel(const int* __restrict__ x, const float* __restrict__ emb,
                                  float* __restrict__ h) {
    const int t = blockIdx.x * blockDim.x + threadIdx.x;  // over NTOT*H
    const int node = t >> 7;
    const int d = t & 127;
    h[t] = emb[(long)(x[node] - 1) * H + d];
}

__global__ void zero_f32_kernel(float* __restrict__ p, long n) {
    const long t = (long)blockIdx.x * blockDim.x + threadIdx.x;
    if (t < n) p[t] = 0.0f;
}

// m_in[dst] += w_in * h_in[src] ; m_out[src] += w_out * h_out[dst]
// m is N x 2H row-major: [:, 0:H] = m_in, [:, H:2H] = m_out
__global__ void edge_scatter_kernel(const int* __restrict__ ei, const float* __restrict__ ec,
                                    const float* __restrict__ idi, const float* __restrict__ odi,
                                    const float* __restrict__ h_in, const float* __restrict__ h_out,
                                    float* __restrict__ m) {
    const int t = blockIdx.x * blockDim.x + threadIdx.x;  // over NE*H
    const int e = t >> 7;
    const int d = t & 127;
    const int src = ei[e];
    const int dst = ei[NE + e];
    const float w_in = ec[e] * idi[e];
    const float w_out = ec[e] * odi[e];
    atomicAdd(&m[(long)dst * (2 * H) + d], w_in * h_in[(long)src * H + d]);
    atomicAdd(&m[(long)src * (2 * H) + H + d], w_out * h_out[(long)dst * H + d]);
}

// GRU cell elementwise update (a = Wih@m + bih already, g = Whh@h + bhh already)
__global__ void gru_update_kernel(const float* __restrict__ a, const float* __restrict__ g,
                                  const float* __restrict__ h, float* __restrict__ hn) {
    const int t = blockIdx.x * blockDim.x + threadIdx.x;  // over NTOT*H
    const int i = t >> 7;
    const int d = t & 127;
    const float* ai = a + (long)i * (3 * H);
    const float* gi = g + (long)i * (3 * H);
    const float r = sigmoidf_(ai[d] + gi[d]);
    const float z = sigmoidf_(ai[H + d] + gi[H + d]);
    const float nn = tanhf(ai[2 * H + d] + r * gi[2 * H + d]);
    hn[t] = (1.0f - z) * nn + z * h[t];
}

// alpha[i] = dot(sigmoid(vW[batch[i],:] + t1[i,:]), q_w) + q_b   (block = 128 thr / node)
__global__ void alpha_kernel(const float* __restrict__ vW, const float* __restrict__ t1,
                             const float* __restrict__ qw, const float* __restrict__ qb,
                             const int* __restrict__ batch, float* __restrict__ alpha) {
    __shared__ float red[H];
    const int i = blockIdx.x;
    const int d = threadIdx.x;
    const int b = batch[i];
    red[d] = sigmoidf_(vW[(long)b * H + d] + t1[(long)i * H + d]) * qw[d];
    __syncthreads();
    for (int off = H / 2; off > 0; off >>= 1) {
        if (d < off) red[d] += red[d + off];
        __syncthreads();
    }
    if (d == 0) alpha[i] = red[0] + qb[0];
}

// cat[b] = [ v_n(b) , sum_j num_count*alpha*hn ]   (block = 128 thr / graph)
__global__ void readout_kernel(const float* __restrict__ hn, const float* __restrict__ alpha,
                               const float* __restrict__ num_count, float* __restrict__ cat) {
    const int b = blockIdx.x;
    const int d = threadIdx.x;
    float acc = 0.0f;
#pragma unroll 4
    for (int j = 0; j < NPER; ++j) {
        const int i = b * NPER + j;
        acc += num_count[i] * alpha[i] * hn[(long)i * H + d];
    }
    cat[(long)b * (2 * H) + d] = hn[(long)(b * NPER + NPER - 1) * H + d];  // v_n (last node)
    cat[(long)b * (2 * H) + H + d] = acc;                                  // s_g
}

static inline int gemm_blocks(int M, int N) { return ((M >> 4) * (N >> 4) + 7) / 8; }

extern "C" void kernel_launch(void* const* d_in, const int* in_sizes, int n_in,
                              void* d_out, int out_size, void* d_ws, size_t ws_size,
                              hipStream_t stream) {
    (void)in_sizes; (void)n_in; (void)out_size; (void)ws_size;

    const int*   x          = (const int*)d_in[0];
    const int*   edge_index = (const int*)d_in[1];
    const int*   batch      = (const int*)d_in[2];
    const float* edge_count = (const float*)d_in[3];
    const float* in_deg     = (const float*)d_in[4];
    const float* out_deg    = (const float*)d_in[5];
    const float* num_count  = (const float*)d_in[6];
    /* d_in[7] = num_graphs scalar (compile-time NB here) */
    const float* emb     = (const float*)d_in[8];
    const float* ggnn_w  = (const float*)d_in[9];
    const float* ggnn_b  = (const float*)d_in[10];
    const float* gru_wih = (const float*)d_in[11];
    const float* gru_whh = (const float*)d_in[12];
    const float* gru_bih = (const float*)d_in[13];
    const float* gru_bhh = (const float*)d_in[14];
    const float* W1_w = (const float*)d_in[15];
    const float* W1_b = (const float*)d_in[16];
    const float* W2_w = (const float*)d_in[17];
    const float* W2_b = (const float*)d_in[18];
    const float* q_w  = (const float*)d_in[19];
    const float* q_b  = (const float*)d_in[20];
    const float* W3_w = (const float*)d_in[21];
    const float* W3_b = (const float*)d_in[22];
    float* out = (float*)d_out;
    float* ws  = (float*)d_ws;

    const long NH = (long)NTOT * H;  // 4,194,304 floats
    float* h    = ws;            // [N,H]
    float* hin  = ws + 1 * NH;   // [N,H]  (reused as hn after scatter)
    float* hout = ws + 2 * NH;   // [N,H]  (reused as t1 after scatter)
    float* m    = ws + 3 * NH;   // [N,2H]
    float* a    = ws + 5 * NH;   // [N,3H]
    float* g    = ws + 8 * NH;   // [N,3H]
    float* hn   = hin;
    float* t1   = hout;
    float* vW    = ws + 11 * NH;             // [B,H]
    float* alpha = vW + (long)NB * H;        // [N]
    float* cat   = alpha + NTOT;             // [B,2H]
    float* s_h   = cat + (long)NB * 2 * H;   // [B,H]

    const int TPB = 256;

    // 1) h = emb[x-1]
    gather_emb_kernel<<<(int)(NH / TPB), TPB, 0, stream>>>(x, emb, h);

    // 2) h_in = h @ ggnn_w[0] + ggnn_b[0] ; h_out = h @ ggnn_w[1] + ggnn_b[1]
    gemm_wmma_f32<false><<<gemm_blocks(NTOT, H), TPB, 0, stream>>>(
        h, ggnn_w, ggnn_b, hin, NTOT, H, H, H, H, H);
    gemm_wmma_f32<false><<<gemm_blocks(NTOT, H), TPB, 0, stream>>>(
        h, ggnn_w + H * H, ggnn_b + H, hout, NTOT, H, H, H, H, H);

    // 3) m = segment sums of weighted messages (atomic scatter over edges)
    zero_f32_kernel<<<(int)((2 * NH + TPB - 1) / TPB), TPB, 0, stream>>>(m, 2 * NH);
    edge_scatter_kernel<<<(NE * H) / TPB, TPB, 0, stream>>>(
        edge_index, edge_count, in_deg, out_deg, hin, hout, m);

    // 4) a = m @ gru_wih^T + bih ; g = h @ gru_whh^T + bhh ; GRU elementwise
    gemm_wmma_f32<true><<<gemm_blocks(NTOT, 3 * H), TPB, 0, stream>>>(
        m, gru_wih, gru_bih, a, NTOT, 3 * H, 2 * H, 2 * H, 2 * H, 3 * H);
    gemm_wmma_f32<true><<<gemm_blocks(NTOT, 3 * H), TPB, 0, stream>>>(
        h, gru_whh, gru_bhh, g, NTOT, 3 * H, H, H, H, 3 * H);
    gru_update_kernel<<<(int)(NH / TPB), TPB, 0, stream>>>(a, g, h, hn);

    // 5) attention readout
    //    t1 = hn @ W2^T + b2   [N,H]
    gemm_wmma_f32<true><<<gemm_blocks(NTOT, H), TPB, 0, stream>>>(
        hn, W2_w, W2_b, t1, NTOT, H, H, H, H, H);
    //    vW = v_n @ W1^T + b1  [B,H]; v_n rows = hn[b*NPER+31] via lda stride trick
    gemm_wmma_f32<true><<<gemm_blocks(NB, H), TPB, 0, stream>>>(
        hn + (NPER - 1) * H, W1_w, W1_b, vW, NB, H, H, NPER * H, H, H);
    alpha_kernel<<<NTOT, H, 0, stream>>>(vW, t1, q_w, q_b, batch, alpha);
    readout_kernel<<<NB, H, 0, stream>>>(hn, alpha, num_count, cat);
    //    s_h = cat @ W3^T + b3  [B,H]
    gemm_wmma_f32<true><<<gemm_blocks(NB, H), TPB, 0, stream>>>(
        cat, W3_w, W3_b, s_h, NB, H, 2 * H, 2 * H, 2 * H, H);

    // 6) scores = s_h @ emb^T  [B, NNODE] — register-blocked 16x80 wave tiles
    score_gemm_wmma<<<SCORE_WAVES / 8, TPB, 0, stream>>>(s_h, emb, out);
}